// ModelScrewDriver_87273735454920
// MI455X (gfx1250) — compile-verified
//
#include <hip/hip_runtime.h>
#include <math.h>

// ---- problem constants ----
#define NB    8
#define NL    24
#define NR    64
#define NDL   4096
#define NFUSE 1536
#define NDT   18432   // 2*12*768
#define NDP   4096
// output layout in d_out (floats): A_large | B_large | gate
#define OFF_A   0
#define OFF_B   50331648   // 8*24*64*4096
#define OFF_G   100663296  // OFF_B + 8*24*4096*64

// ---- workspace layout (floats); total = 942272 floats = 3.77 MB ----
#define WS_FUSED 0                       // [16][1536]  (rows 8..15 zero)
#define WS_FA    24576                   // [16][4096]
#define WS_FB    90112                   // [16][4096]
#define WS_LA    155648                  // [32][4096]  (rows 24..31 unused)
#define WS_LB    286720                  // [32][4096]
#define WS_CA    417792                  // [64][4096]  C_A[r][d]
#define WS_CBT   679936                  // [4096][64]  C_B transposed: [d][r]
#define WS_GATE  942080                  // [192]
#define WS_ZROW  (WS_FUSED + 8 * NFUSE)  // guaranteed-zero row (fused pad row 8)

typedef __attribute__((ext_vector_type(2))) float v2f;
typedef __attribute__((ext_vector_type(4))) float v4f;
typedef __attribute__((ext_vector_type(8))) float v8f;

// ---------------------------------------------------------------------------
// K1: fused = [relu(flat_T @ W_ct + b_ct) | relu(prompt @ W_cp + b_cp)], padded
// to 16 rows.  Blocks 0..63 -> t_feat cols (16 each), 64..95 -> p_feat cols.
// 256 threads = 16 cols x 16 k-lanes; LDS reduction over the 16 k-partials.
// ---------------------------------------------------------------------------
__global__ void k_fused(const float* __restrict__ As, const float* __restrict__ Bs,
                        const float* __restrict__ prompt,
                        const float* __restrict__ W_ct, const float* __restrict__ b_ct,
                        const float* __restrict__ W_cp, const float* __restrict__ b_cp,
                        float* __restrict__ ws) {
  __shared__ float red[16][16][NB];          // [kk][cc][b]
  const int tid = threadIdx.x;
  const int cc = tid & 15, kk = tid >> 4;
  const int blk = blockIdx.x;

  float acc[NB];
#pragma unroll
  for (int b = 0; b < NB; ++b) acc[b] = 0.f;

  if (blk < 64) {                            // t_feat, fused cols [0,1024)
    const int col = blk * 16 + cc;
    for (int k = kk; k < 9216; k += 16) {    // A_small half of flat_T
      float w = __builtin_nontemporal_load(&W_ct[k * 1024 + col]);
#pragma unroll
      for (int b = 0; b < NB; ++b) acc[b] = fmaf(As[b * 9216 + k], w, acc[b]);
    }
    for (int k = kk; k < 9216; k += 16) {    // B_small half of flat_T
      float w = __builtin_nontemporal_load(&W_ct[(k + 9216) * 1024 + col]);
#pragma unroll
      for (int b = 0; b < NB; ++b) acc[b] = fmaf(Bs[b * 9216 + k], w, acc[b]);
    }
  } else {                                   // p_feat, fused cols [1024,1536)
    const int colL = (blk - 64) * 16 + cc;
    for (int k = kk; k < NDP; k += 16) {
      float w = __builtin_nontemporal_load(&W_cp[k * 512 + colL]);
#pragma unroll
      for (int b = 0; b < NB; ++b) acc[b] = fmaf(prompt[b * NDP + k], w, acc[b]);
    }
  }
#pragma unroll
  for (int b = 0; b < NB; ++b) red[kk][cc][b] = acc[b];
  __syncthreads();

  if (tid < 128) {                           // 16 cols x 8 batches
    const int c2 = tid & 15, b = tid >> 4;
    float s = 0.f;
#pragma unroll
    for (int j = 0; j < 16; ++j) s += red[j][c2][b];
    int colF;
    if (blk < 64) { colF = blk * 16 + c2; s += b_ct[colF]; }
    else { int colL = (blk - 64) * 16 + c2; colF = 1024 + colL; s += b_cp[colL]; }
    ws[WS_FUSED + b * NFUSE + colF] = fmaxf(s, 0.f);
  } else {                                   // zero pad rows 8..15
    const int t = tid - 128;
    const int c2 = t & 15, row = 8 + (t >> 4);
    const int colF = (blk < 64) ? blk * 16 + c2 : 1024 + (blk - 64) * 16 + c2;
    ws[WS_FUSED + row * NFUSE + colF] = 0.f;
  }
}

// ---------------------------------------------------------------------------
// K2: gate = hard(sigmoid(fused @ W_pc + b_pc) > 0.5)  (straight-through fwd)
// ---------------------------------------------------------------------------
__global__ void k_gate(const float* __restrict__ W_pc, const float* __restrict__ b_pc,
                       float* __restrict__ ws, float* __restrict__ out) {
  const int t = threadIdx.x;
  if (t >= NB * NL) return;
  const int b = t / NL, l = t % NL;
  const float* f = ws + WS_FUSED + b * NFUSE;
  float s = b_pc[l];
  for (int k = 0; k < NFUSE; ++k) s = fmaf(f[k], W_pc[k * NL + l], s);
  const float g = (s > 0.f) ? 1.f : 0.f;     // sigmoid(s) > 0.5  <=>  s > 0
  ws[WS_GATE + t] = g;
  out[OFF_G + t] = g;
}

// ---------------------------------------------------------------------------
// K3: basis GEMMs via V_WMMA_F32_16X16X4_F32, one 16x16 tile per wave.
// 14 row-tiles x 256 col-tiles = 3584 waves (448 blocks x 8 waves).
//   rt 0/1   : F_A/F_B  = fused_pad[16,1536] @ W[0:1536]
//   rt 2..5  : L_A/L_B  = E_layer_pad[32,256] @ W[1536:1792]   (2 tiles each)
//   rt 6..13 : C_A/C_B  = E_chunk[64,64]      @ W[1792:1856]   (4 tiles each;
//              C_B stored transposed [d][r])
// Padded rows (E_layer m>=24) read from a guaranteed-zero ws row instead of
// predicating the load, so the inner loop is branch-free and EXEC stays all-1.
// ---------------------------------------------------------------------------
__global__ void k_wmma(const float* __restrict__ E_layer, const float* __restrict__ E_chunk,
                       const float* __restrict__ W_A, const float* __restrict__ W_B,
                       float* __restrict__ ws) {
  const int wave = blockIdx.x * 8 + (threadIdx.x >> 5);
  const int lane = threadIdx.x & 31;
  const int hi = lane >> 4, lo = lane & 15;
  const int col0 = (wave & 255) * 16;
  const int rt = wave >> 8;

  const float* Aptr; const float* Wm; float* outp;
  int lda, K, krow0, mbase, Mvalid;
  bool transposedStore = false;

  if (rt < 2) {
    Wm = (rt == 0) ? W_A : W_B;
    outp = ws + ((rt == 0) ? WS_FA : WS_FB);
    Aptr = ws + WS_FUSED; lda = NFUSE; K = NFUSE; krow0 = 0; mbase = 0; Mvalid = 16;
  } else if (rt < 6) {
    const int q = rt - 2;                    // 0,1 -> A tiles ; 2,3 -> B tiles
    const int tile = q & 1;
    Wm = (q < 2) ? W_A : W_B;
    outp = ws + ((q < 2) ? WS_LA : WS_LB) + tile * 16 * NDL;
    Aptr = E_layer; lda = 256; K = 256; krow0 = NFUSE; mbase = tile * 16; Mvalid = NL;
  } else {
    const int q = rt - 6;                    // 0..3 -> C_A tiles ; 4..7 -> C_B
    const int tile = q & 3;
    const bool isB = q >= 4;
    Wm = isB ? W_B : W_A;
    Aptr = E_chunk; lda = 64; K = 64; krow0 = NFUSE + 256; mbase = tile * 16; Mvalid = NR;
    transposedStore = isB;
    outp = isB ? (ws + WS_CBT) : (ws + WS_CA + tile * 16 * NDL);
  }

  const int m = mbase + lo;
  // Invalid rows read a zero row (fused pad row 8, 1536 zeros >= any K here).
  const float* arow = (m < Mvalid) ? (Aptr + m * lda) : (ws + WS_ZROW);
  const float* brow = Wm + (size_t)(krow0 + 2 * hi) * NDL + col0 + lo;

  v8f acc = {0.f, 0.f, 0.f, 0.f, 0.f, 0.f, 0.f, 0.f};

#if __has_builtin(__builtin_amdgcn_wmma_f32_16x16x4_f32)
#pragma unroll 4
  for (int k = 0; k < K; k += 4) {
    // A frag (ISA 32-bit 16x4 layout): lanes 0-15 hold K=k,k+1 ; 16-31 K=k+2,k+3
    v2f a = *(const v2f*)(arow + k + 2 * hi);
    // B frag (4x16): rows k+2*hi and k+2*hi+1, col = lane&15
    v2f bfrag;
    bfrag.x = brow[(size_t)k * NDL];
    bfrag.y = brow[(size_t)(k + 1) * NDL];
    acc = __builtin_amdgcn_wmma_f32_16x16x4_f32(false, a, false, bfrag,
                                                (short)0, acc, false, false);
  }
#else
  // scalar fallback: lane owns D rows (i + 8*hi), col = lo
#pragma unroll
  for (int i = 0; i < 8; ++i) {
    const int mg = mbase + i + 8 * hi;
    float s = 0.f;
    if (mg < Mvalid)
      for (int k = 0; k < K; ++k)
        s = fmaf(Aptr[mg * lda + k], Wm[(size_t)(krow0 + k) * NDL + col0 + lo], s);
    acc[i] = s;
  }
#endif

  if (transposedStore) {                     // C_BT[d][r]
#pragma unroll
    for (int i = 0; i < 8; ++i)
      outp[(col0 + lo) * 64 + (mbase + i + 8 * hi)] = acc[i];
  } else {                                   // row-major [m][4096] (mbase folded)
#pragma unroll
    for (int i = 0; i < 8; ++i)
      outp[(i + 8 * hi) * NDL + col0 + lo] = acc[i];
  }
}

// ---------------------------------------------------------------------------
// K4: broadcast-add writer (the 402 MB bandwidth kernel).
//   blocks [0,12288)     : A_large[b,l,r,:] = (F_A[b]+L_A[l]+C_A[r]+b_A)*gate
//   blocks [12288,24576) : B_large[b,l,d0:d0+64,:] = F_B[b,d]+L_B[l,d]+b_B[d]+C_BT[d][r]
// Each block writes one contiguous 16 KB chunk with nontemporal b128 stores.
// Basis tables (~3 MB) stay hot in the 192 MB L2, so DRAM traffic ~= writes.
// ---------------------------------------------------------------------------
__global__ void k_out(const float* __restrict__ b_A, const float* __restrict__ b_B,
                      const float* __restrict__ ws, float* __restrict__ out) {
  const int blk = blockIdx.x;
  const int tid = threadIdx.x;
  if (blk < NB * NL * NR) {
    const int b = blk / (NL * NR), rem = blk % (NL * NR), l = rem / NR, r = rem % NR;
    const float g = ws[WS_GATE + b * NL + l];
    const float* fa = ws + WS_FA + b * NDL;
    const float* la = ws + WS_LA + l * NDL;
    const float* ca = ws + WS_CA + r * NDL;
    float* o = out + (size_t)blk * NDL;
#pragma unroll
    for (int p = 0; p < 4; ++p) {
      const int d = p * 1024 + tid * 4;
      v4f vf = *(const v4f*)(fa + d);
      v4f vl = *(const v4f*)(la + d);
      v4f vc = *(const v4f*)(ca + d);
      v4f vb = *(const v4f*)(b_A + d);
      v4f v = (vf + vl + vc + vb) * g;
      __builtin_nontemporal_store(v, (v4f*)(o + d));
    }
  } else {
    const int j = blk - NB * NL * NR;
    const int bl = j >> 6;                   // (b*24+l)
    const int d0 = (j & 63) * 64;
    const int b = bl / NL, l = bl % NL;
    const float* fb = ws + WS_FB + b * NDL;
    const float* lb = ws + WS_LB + l * NDL;
    float* o = out + (size_t)OFF_B + ((size_t)bl * NDL + d0) * NR;
#pragma unroll
    for (int p = 0; p < 4; ++p) {
      const int dl = p * 16 + (tid >> 4);    // local d in [0,64)
      const int d = d0 + dl;
      const int r = (tid & 15) * 4;
      const float s = fb[d] + lb[d] + b_B[d];
      v4f vc = *(const v4f*)(ws + WS_CBT + d * 64 + r);
      v4f v = vc + s;
      __builtin_nontemporal_store(v, (v4f*)(o + dl * 64 + r));
    }
  }
}

// ---------------------------------------------------------------------------
extern "C" void kernel_launch(void* const* d_in, const int* in_sizes, int n_in,
                              void* d_out, int out_size, void* d_ws, size_t ws_size,
                              hipStream_t stream) {
  const float* A_small = (const float*)d_in[0];
  const float* B_small = (const float*)d_in[1];
  const float* prompt  = (const float*)d_in[2];
  const float* W_ct    = (const float*)d_in[3];
  const float* b_ct    = (const float*)d_in[4];
  const float* W_cp    = (const float*)d_in[5];
  const float* b_cp    = (const float*)d_in[6];
  const float* W_pc    = (const float*)d_in[7];
  const float* b_pc    = (const float*)d_in[8];
  const float* E_layer = (const float*)d_in[9];
  const float* E_chunk = (const float*)d_in[10];
  const float* W_A     = (const float*)d_in[11];
  const float* b_A     = (const float*)d_in[12];
  const float* W_B     = (const float*)d_in[13];
  const float* b_B     = (const float*)d_in[14];
  float* out = (float*)d_out;
  float* ws  = (float*)d_ws;                 // needs >= 3.77 MB

  k_fused<<<96, 256, 0, stream>>>(A_small, B_small, prompt, W_ct, b_ct, W_cp, b_cp, ws);
  k_gate<<<1, 192, 0, stream>>>(W_pc, b_pc, ws, out);
  k_wmma<<<448, 256, 0, stream>>>(E_layer, E_chunk, W_A, W_B, ws);
  k_out<<<24576, 256, 0, stream>>>(b_A, b_B, ws, out);
}